// RelinearE_14920716386929
// MI455X (gfx1250) — compile-verified
//
#include <hip/hip_runtime.h>
#include <math.h>

#define NE_ 14541
#define R2_ 474
#define D_  200
#define TD_ 400
#define B_  512

typedef __attribute__((ext_vector_type(2))) float v2f;
typedef __attribute__((ext_vector_type(8))) float v8f;

// BatchNorm eval scale: 1/sqrt(1 + 1e-5)
#define BN_S 0.9999950000374997f

// ---------------------------------------------------------------------------
// Stage 1: gather + concat  ->  X[b][0:200]=ent_emb[h[b]], X[b][200:400]=rel_emb[r[b]]
// ---------------------------------------------------------------------------
__global__ void gather_concat_kernel(const int* __restrict__ h,
                                     const int* __restrict__ r,
                                     const float* __restrict__ ent,
                                     const float* __restrict__ rel,
                                     float* __restrict__ X) {
    int idx = blockIdx.x * blockDim.x + threadIdx.x;
    if (idx >= B_ * TD_) return;
    int b = idx / TD_;
    int j = idx - b * TD_;
    float v;
    if (j < D_) v = ent[h[b] * D_ + j];
    else        v = rel[r[b] * D_ + (j - D_)];
    X[idx] = v;
}

// ---------------------------------------------------------------------------
// Stage 2: per-relation matvec + bias + BN(eval) + ReLU.
// One workgroup per batch element (pure HBM-bandwidth stage: 640KB matrix per b).
// x vector staged in LDS; each wave owns rows [wave, wave+8, ...]; lanes split
// the K=400 dot product via coalesced float4 loads, then shuffle-reduce.
// ---------------------------------------------------------------------------
__global__ void __launch_bounds__(256)
rel_layer_kernel(const float* __restrict__ W,     // [R2, TD, TD]
                 const float* __restrict__ bvec,  // [R2]
                 const int* __restrict__ r,       // [B]
                 const float* __restrict__ g,     // [TD]
                 const float* __restrict__ be,    // [TD]
                 const float* __restrict__ X,     // [B, TD] in
                 float* __restrict__ Y) {         // [B, TD] out
    __shared__ float xs[TD_];
    const int b   = blockIdx.x;
    const int tid = threadIdx.x;
    for (int i = tid; i < TD_; i += 256) xs[i] = X[b * TD_ + i];
    __syncthreads();

    const int rb = r[b];
    const float* __restrict__ Wr = W + (size_t)rb * (TD_ * TD_);
    const float bias = bvec[rb];
    const int wave = tid >> 5;
    const int lane = tid & 31;

    for (int row = wave; row < TD_; row += 8) {
        const float4* __restrict__ wrow = (const float4*)(Wr + row * TD_);
        float acc = 0.f;
        #pragma unroll
        for (int j4 = lane; j4 < TD_ / 4; j4 += 32) {   // 100 float4 per row
            float4 wv = wrow[j4];
            int j = j4 * 4;
            acc += wv.x * xs[j] + wv.y * xs[j + 1] + wv.z * xs[j + 2] + wv.w * xs[j + 3];
        }
        #pragma unroll
        for (int off = 16; off > 0; off >>= 1)
            acc += __shfl_down(acc, off, 32);
        if (lane == 0) {
            float v = acc + bias;
            v = g[row] * (v * BN_S) + be[row];
            Y[b * TD_ + row] = v > 0.f ? v : 0.f;
        }
    }
}

// ---------------------------------------------------------------------------
// Stage 3: shared linear [512x400]x[400->200] + BN + ReLU via fp32 WMMA 16x16x4.
// A[m][k] = X[m][k]; B[k][n] = tw[n][k] (tw is row-major [D, TD]).
// One wave per 16x16 output tile; 8 waves per block.
// ---------------------------------------------------------------------------
__global__ void __launch_bounds__(256)
linear_bn_relu_wmma(const float* __restrict__ X,   // [B, TD]
                    const float* __restrict__ tw,  // [D, TD]
                    const float* __restrict__ tb,  // [D]
                    const float* __restrict__ g3,  // [D]
                    const float* __restrict__ be3, // [D]
                    float* __restrict__ Z) {       // [B, D]
    const int NT_N = (D_ + 15) / 16;               // 13
    const int wave = threadIdx.x >> 5;
    const int tile = blockIdx.x * 8 + wave;
    const int mt = tile / NT_N;
    const int nt = tile - mt * NT_N;
    if (mt >= B_ / 16) return;                     // wave-uniform

    const int lane = threadIdx.x & 31;
    const int half = lane >> 4;
    const int l15  = lane & 15;
    const int m = mt * 16 + l15;                   // A row (always valid, 512%16==0)
    const int n = nt * 16 + l15;                   // output column
    const bool nvalid = (n < D_);
    const int nc = nvalid ? n : 0;                 // clamped for safe loads

    const float* __restrict__ ap = X  + m  * TD_ + 2 * half;
    const float* __restrict__ bp = tw + nc * TD_ + 2 * half;

    v8f c = {};
    #pragma unroll 4
    for (int k0 = 0; k0 < TD_; k0 += 4) {
        v2f a  = *(const v2f*)(ap + k0);
        v2f bv = *(const v2f*)(bp + k0);
        if (!nvalid) { bv.x = 0.f; bv.y = 0.f; }   // EXEC stays all-ones
        c = __builtin_amdgcn_wmma_f32_16x16x4_f32(false, a, false, bv,
                                                  (short)0, c, false, false);
    }
    if (!nvalid) return;

    const float tbn = tb[n], gn = g3[n], ben = be3[n];
    const int mbase = mt * 16 + 8 * half;
    #pragma unroll
    for (int v = 0; v < 8; ++v) {                  // C: VGPR v -> row v + 8*half
        float val = c[v] + tbn;
        val = gn * (val * BN_S) + ben;
        val = val > 0.f ? val : 0.f;
        Z[(mbase + v) * D_ + n] = val;
    }
}

// ---------------------------------------------------------------------------
// Stage 4: scoring GEMM [512x200]x[200->14541] + bias + sigmoid via fp32 WMMA.
// ent_emb (11.6 MB) lives in L2 (192 MB), so this is compute/L2-bound.
// ---------------------------------------------------------------------------
__global__ void __launch_bounds__(256)
score_sigmoid_wmma(const float* __restrict__ Zm,   // [B, D]
                   const float* __restrict__ ent,  // [NE, D]
                   const float* __restrict__ bias, // [NE]
                   float* __restrict__ out) {      // [B, NE]
    const int NT_N = (NE_ + 15) / 16;              // 909
    const int wave = threadIdx.x >> 5;
    const int tile = blockIdx.x * 8 + wave;
    const int mt = tile / NT_N;
    const int nt = tile - mt * NT_N;
    if (mt >= B_ / 16) return;                     // wave-uniform

    const int lane = threadIdx.x & 31;
    const int half = lane >> 4;
    const int l15  = lane & 15;
    const int m = mt * 16 + l15;
    const int n = nt * 16 + l15;
    const bool nvalid = (n < NE_);
    const int nc = nvalid ? n : 0;

    const float* __restrict__ ap = Zm  + m * D_ + 2 * half;
    const float* __restrict__ bp = ent + (size_t)nc * D_ + 2 * half;

    v8f c = {};
    #pragma unroll 5
    for (int k0 = 0; k0 < D_; k0 += 4) {           // 50 WMMA ops
        v2f a  = *(const v2f*)(ap + k0);
        v2f bv = *(const v2f*)(bp + k0);
        if (!nvalid) { bv.x = 0.f; bv.y = 0.f; }
        c = __builtin_amdgcn_wmma_f32_16x16x4_f32(false, a, false, bv,
                                                  (short)0, c, false, false);
    }
    if (!nvalid) return;

    const float bn = bias[n];
    const int mbase = mt * 16 + 8 * half;
    #pragma unroll
    for (int v = 0; v < 8; ++v) {
        float x = c[v] + bn;
        float p = 1.0f / (1.0f + __expf(-x));
        out[(size_t)(mbase + v) * NE_ + n] = p;
    }
}

// ---------------------------------------------------------------------------
extern "C" void kernel_launch(void* const* d_in, const int* in_sizes, int n_in,
                              void* d_out, int out_size, void* d_ws, size_t ws_size,
                              hipStream_t stream) {
    const int*   h    = (const int*)  d_in[0];
    const int*   r    = (const int*)  d_in[1];
    const float* ent  = (const float*)d_in[2];
    const float* rel  = (const float*)d_in[3];
    const float* w0   = (const float*)d_in[4];
    const float* w1   = (const float*)d_in[5];
    // d_in[6] (w_r2), d_in[9] (b2): unused by the reference
    const float* b0   = (const float*)d_in[7];
    const float* b1   = (const float*)d_in[8];
    const float* tw   = (const float*)d_in[10];
    const float* tb   = (const float*)d_in[11];
    const float* g0   = (const float*)d_in[12];
    const float* be0  = (const float*)d_in[13];
    const float* g1   = (const float*)d_in[14];
    const float* be1  = (const float*)d_in[15];
    const float* g3   = (const float*)d_in[16];
    const float* be3  = (const float*)d_in[17];
    const float* bias = (const float*)d_in[18];

    float* ws = (float*)d_ws;
    float* X0 = ws;                    // [B, TD]
    float* X1 = ws + B_ * TD_;         // [B, TD]
    float* Zb = ws + 2 * B_ * TD_;     // [B, D]
    float* out = (float*)d_out;        // [B, NE]

    gather_concat_kernel<<<(B_ * TD_ + 255) / 256, 256, 0, stream>>>(h, r, ent, rel, X0);

    // layer 0: X0 -> X1 ; layer 1: X1 -> X0 (ping-pong)
    rel_layer_kernel<<<B_, 256, 0, stream>>>(w0, b0, r, g0, be0, X0, X1);
    rel_layer_kernel<<<B_, 256, 0, stream>>>(w1, b1, r, g1, be1, X1, X0);

    // shared linear: 32 M-tiles x 13 N-tiles = 416 tiles, 8 waves/block
    linear_bn_relu_wmma<<<52, 256, 0, stream>>>(X0, tw, tb, g3, be3, Zb);

    // scoring: 32 x 909 = 29088 tiles, 8 waves/block
    score_sigmoid_wmma<<<3636, 256, 0, stream>>>(Zb, ent, bias, out);
}